// AttentionC_34419867910990
// MI455X (gfx1250) — compile-verified
//
#include <hip/hip_runtime.h>
#include <math.h>

#define DIM    192
#define HEADS  8
#define HD     24          // head_dim
#define NTOK   32768       // 8*64*64
#define BATCH  2
#define CQKV   576         // 3*DIM

typedef float v2f __attribute__((ext_vector_type(2)));
typedef float v8f __attribute__((ext_vector_type(8)));

// D = A(16x4 f32) * B(4x16 f32) + C(16x16 f32), full fp32 matrix pipe.
__device__ __forceinline__ v8f wmma_f32_k4(v2f a, v2f b, v8f c) {
    return __builtin_amdgcn_wmma_f32_16x16x4_f32(
        /*neg_a=*/false, a, /*neg_b=*/false, b,
        /*c_mod=*/(short)0, c, /*reuse_a=*/false, /*reuse_b=*/false);
}

// ---------------------------------------------------------------------------
// Y[b,M,N] = W[M,K] * X[b,K,N] + bias  (pointwise conv as GEMM)
// Block = 128 threads = 4 waves. Each wave owns a 16(M) x 64(N) tile:
// one A (weights) float2 load feeds 4 independent WMMA accumulator chains.
// A layout (16x4 f32): lane l holds row M=l%16, K = 2*(l/16)+{0,1}.
// B layout (4x16 f32): lane l holds col N=l%16, K = 2*(l/16)+{vgpr0,vgpr1}.
// ---------------------------------------------------------------------------
template <int M, int K>
__global__ __launch_bounds__(128) void gemm_bias_kernel(
    const float* __restrict__ X,     // [B, K, NTOK]
    const float* __restrict__ W,     // [M, K]
    const float* __restrict__ bias,  // [M]
    float* __restrict__ Y)           // [B, M, NTOK]
{
    const int N    = NTOK;
    const int lane = threadIdx.x & 31;
    const int wave = threadIdx.x >> 5;
    const int n0   = blockIdx.x * 64;
    const int m0   = (blockIdx.y * 4 + wave) * 16;
    const int b    = blockIdx.z;

    const float* __restrict__ Xb = X + (size_t)b * K * N;
    float* __restrict__       Yb = Y + (size_t)b * M * N;

    const int mrow  = m0 + (lane & 15);
    const int khalf = (lane >> 4) * 2;      // 0 or 2
    const int col   = lane & 15;

    const float* __restrict__ wrow  = W + (size_t)mrow * K + khalf;
    const float* __restrict__ xbase = Xb + (size_t)khalf * N + n0 + col;

    v8f acc0 = {}, acc1 = {}, acc2 = {}, acc3 = {};
    for (int k0 = 0; k0 < K; k0 += 4) {
        v2f a = *(const v2f*)(wrow + k0);            // contiguous, 8B aligned
        const float* __restrict__ xp = xbase + (size_t)k0 * N;
        v2f b0, b1, b2, b3;
        b0.x = xp[0];  b0.y = xp[N];
        b1.x = xp[16]; b1.y = xp[N + 16];
        b2.x = xp[32]; b2.y = xp[N + 32];
        b3.x = xp[48]; b3.y = xp[N + 48];
        acc0 = wmma_f32_k4(a, b0, acc0);
        acc1 = wmma_f32_k4(a, b1, acc1);
        acc2 = wmma_f32_k4(a, b2, acc2);
        acc3 = wmma_f32_k4(a, b3, acc3);
    }

    // C/D layout: vgpr r, lanes 0-15: M=r ; lanes 16-31: M=8+r ; N = lane&15
    const int rsh = 8 * (lane >> 4);
    #pragma unroll
    for (int r = 0; r < 8; ++r) {
        const int m  = m0 + r + rsh;
        const float bm = bias[m];
        float* __restrict__ yp = Yb + (size_t)m * N + n0 + col;
        yp[0]  = acc0[r] + bm;
        yp[16] = acc1[r] + bm;
        yp[32] = acc2[r] + bm;
        yp[48] = acc3[r] + bm;
    }
}

// ---------------------------------------------------------------------------
// Depthwise 3x3x3, pad 1, on [B, 576, 8, 64, 64]. Interior fast path
// (~72% of elements) runs 27 unguarded FMAs.
// ---------------------------------------------------------------------------
__global__ __launch_bounds__(256) void dwconv_kernel(
    const float* __restrict__ in,    // [B,576,8,64,64]
    const float* __restrict__ w,     // [576,27]
    const float* __restrict__ bias,  // [576]
    float* __restrict__ out)
{
    const int T = 8, H = 64, W = 64;
    size_t idx = (size_t)blockIdx.x * 256 + threadIdx.x;
    const size_t total = (size_t)BATCH * CQKV * T * H * W;
    if (idx >= total) return;

    int x = (int)(idx % W); size_t t1 = idx / W;
    int y = (int)(t1 % H);  t1 /= H;
    int t = (int)(t1 % T);  t1 /= T;
    int c = (int)(t1 % CQKV);
    int b = (int)(t1 / CQKV);

    const float* __restrict__ ip = in + ((size_t)b * CQKV + c) * (T * H * W);
    const float* __restrict__ wp = w + (size_t)c * 27;

    float acc = bias[c];
    if (t >= 1 && t <= 6 && y >= 1 && y <= 62 && x >= 1 && x <= 62) {
        const float* __restrict__ p0 = ip + ((size_t)(t - 1) * H + (y - 1)) * W + (x - 1);
        #pragma unroll
        for (int kt = 0; kt < 3; ++kt) {
            #pragma unroll
            for (int ky = 0; ky < 3; ++ky) {
                const float* __restrict__ row = p0 + (size_t)kt * H * W + (size_t)ky * W;
                #pragma unroll
                for (int kx = 0; kx < 3; ++kx)
                    acc = fmaf(wp[kt * 9 + ky * 3 + kx], row[kx], acc);
            }
        }
    } else {
        #pragma unroll
        for (int kt = 0; kt < 3; ++kt) {
            const int tt = t + kt - 1;
            if (tt < 0 || tt >= T) continue;
            #pragma unroll
            for (int ky = 0; ky < 3; ++ky) {
                const int yy = y + ky - 1;
                if (yy < 0 || yy >= H) continue;
                #pragma unroll
                for (int kx = 0; kx < 3; ++kx) {
                    const int xx = x + kx - 1;
                    if (xx < 0 || xx >= W) continue;
                    acc = fmaf(wp[kt * 9 + ky * 3 + kx],
                               ip[((size_t)tt * H + yy) * W + xx], acc);
                }
            }
        }
    }
    out[idx] = acc;
}

// ---------------------------------------------------------------------------
// 1/max(||row||2, 1e-12) for q (ch 0..191) and k (ch 192..383) rows.
// ---------------------------------------------------------------------------
__global__ __launch_bounds__(256) void rownorm_kernel(
    const float* __restrict__ QKV,   // [B,576,NTOK]
    float* __restrict__ scales)      // [B*384]
{
    const int N = NTOK;
    const int r = blockIdx.x;                  // 0 .. B*384-1
    const int b = r / 384, c = r % 384;
    const float* __restrict__ row = QKV + ((size_t)b * CQKV + c) * N;

    float s = 0.f;
    for (int i = threadIdx.x; i < N / 4; i += 256) {
        float4 v = ((const float4*)row)[i];
        s += v.x * v.x + v.y * v.y + v.z * v.z + v.w * v.w;
    }
    __shared__ float red[256];
    red[threadIdx.x] = s;
    __syncthreads();
    for (int st = 128; st > 0; st >>= 1) {
        if (threadIdx.x < st) red[threadIdx.x] += red[threadIdx.x + st];
        __syncthreads();
    }
    if (threadIdx.x == 0)
        scales[r] = 1.0f / fmaxf(sqrtf(red[0]), 1e-12f);
}

__global__ void zero_kernel(float* __restrict__ p, int count) {
    int i = blockIdx.x * 256 + threadIdx.x;
    if (i < count) p[i] = 0.f;
}

// ---------------------------------------------------------------------------
// Partial q.kT : per (b,h) a 24x24 Gram matrix over K=32768 tokens, padded to
// 32x32. K split across 64 blocks, branch-free fp32 WMMA (padding rows/cols
// read in-bounds neighbor channels and only pollute padding outputs, which
// are never consumed), two independent accumulator chains for latency hiding,
// partials merged with native f32 atomics.
// ---------------------------------------------------------------------------
__global__ __launch_bounds__(128) void qk_partial_kernel(
    const float* __restrict__ QKV,   // [B,576,NTOK] (post depthwise)
    float* __restrict__ accum)       // [B*HEADS, 32, 32]
{
    const int N    = NTOK;
    const int lane = threadIdx.x & 31;
    const int wave = threadIdx.x >> 5;       // 0..3 -> 2x2 tile grid
    const int mt   = wave >> 1, nt = wave & 1;
    const int bh   = blockIdx.z;
    const int b    = bh >> 3, h = bh & 7;
    const int kbeg = blockIdx.x * (N / 64);  // 512-token slab
    const int kend = kbeg + (N / 64);

    const float* __restrict__ qbase = QKV + (size_t)b * CQKV * N;
    const float* __restrict__ kbase = qbase + (size_t)DIM * N;

    const int mrow = mt * 16 + (lane & 15);  // 0..31 (>=24 is padding, in-bounds)
    const int ncol = nt * 16 + (lane & 15);
    const int kh2  = (lane >> 4) * 2;

    const float* __restrict__ aptr = qbase + (size_t)(h * HD + mrow) * N + kh2;
    const float* __restrict__ bptr = kbase + (size_t)(h * HD + ncol) * N + kh2;

    v8f acc0 = {}, acc1 = {};
    for (int k0 = kbeg; k0 < kend; k0 += 8) {
        v2f a0 = *(const v2f*)(aptr + k0);
        v2f b0 = *(const v2f*)(bptr + k0);
        v2f a1 = *(const v2f*)(aptr + k0 + 4);
        v2f b1 = *(const v2f*)(bptr + k0 + 4);
        acc0 = wmma_f32_k4(a0, b0, acc0);
        acc1 = wmma_f32_k4(a1, b1, acc1);
    }
    v8f acc = acc0 + acc1;

    float* __restrict__ dst = accum + (size_t)bh * 32 * 32;
    const int rsh = 8 * (lane >> 4);
    #pragma unroll
    for (int r = 0; r < 8; ++r) {
        const int m = mt * 16 + r + rsh;
        atomicAdd(dst + m * 32 + nt * 16 + (lane & 15), acc[r]);
    }
}

// ---------------------------------------------------------------------------
// attn = softmax( accum * s_q[c] * s_k[d] * temperature[h] ), 24x24 per (b,h)
// ---------------------------------------------------------------------------
__global__ __launch_bounds__(32) void softmax_kernel(
    const float* __restrict__ accum,   // [B*HEADS,32,32]
    const float* __restrict__ scales,  // [B*384]
    const float* __restrict__ temp,    // [HEADS]
    float* __restrict__ attn)          // [B*HEADS,24,24]
{
    const int bh = blockIdx.x;
    const int b  = bh >> 3, h = bh & 7;
    const int c  = threadIdx.x;
    if (c >= HD) return;

    const float* __restrict__ arow = accum + ((size_t)bh * 32 + c) * 32;
    const float sq  = scales[b * 384 + h * HD + c];
    const float tpr = temp[h];

    float vals[HD];
    float mx = -INFINITY;
    #pragma unroll
    for (int d = 0; d < HD; ++d) {
        float v = arow[d] * sq * scales[b * 384 + DIM + h * HD + d] * tpr;
        vals[d] = v;
        mx = fmaxf(mx, v);
    }
    float sum = 0.f;
    #pragma unroll
    for (int d = 0; d < HD; ++d) {
        vals[d] = expf(vals[d] - mx);
        sum += vals[d];
    }
    const float inv = 1.f / sum;
    float* __restrict__ orow = attn + ((size_t)bh * HD + c) * HD;
    #pragma unroll
    for (int d = 0; d < HD; ++d) orow[d] = vals[d] * inv;
}

// ---------------------------------------------------------------------------
// out[b, h*24+c, :] = sum_d attn[b,h,c,d] * v[b, 384+h*24+d, :]
// ---------------------------------------------------------------------------
__global__ __launch_bounds__(256) void av_kernel(
    const float* __restrict__ QKV,   // [B,576,NTOK]
    const float* __restrict__ attn,  // [B*HEADS,24,24]
    float* __restrict__ out)         // [B,192,NTOK]
{
    const int N   = NTOK;
    const int pos = blockIdx.x * 256 + threadIdx.x;
    const int c   = blockIdx.y;       // 0..191
    const int b   = blockIdx.z;
    const int h   = c / HD, ci = c % HD;

    __shared__ float arow[HD];
    if (threadIdx.x < HD)
        arow[threadIdx.x] = attn[(((size_t)b * HEADS + h) * HD + ci) * HD + threadIdx.x];
    __syncthreads();

    const float* __restrict__ vbase =
        QKV + (size_t)b * CQKV * N + (size_t)(2 * DIM + h * HD) * N;
    float acc = 0.f;
    #pragma unroll
    for (int d = 0; d < HD; ++d)
        acc = fmaf(arow[d], vbase[(size_t)d * N + pos], acc);

    out[((size_t)b * DIM + c) * N + pos] = acc;
}

// ---------------------------------------------------------------------------
extern "C" void kernel_launch(void* const* d_in, const int* in_sizes, int n_in,
                              void* d_out, int out_size, void* d_ws, size_t ws_size,
                              hipStream_t stream)
{
    (void)in_sizes; (void)n_in; (void)out_size; (void)ws_size;

    const float* x      = (const float*)d_in[0];
    const float* qkv_w  = (const float*)d_in[1];   // [576,192]
    const float* qkv_b  = (const float*)d_in[2];   // [576]
    const float* dw_w   = (const float*)d_in[3];   // [576,27]
    const float* dw_b   = (const float*)d_in[4];   // [576]
    const float* temp   = (const float*)d_in[5];   // [8]
    const float* proj_w = (const float*)d_in[6];   // [192,192]
    const float* proj_b = (const float*)d_in[7];   // [192]
    float* out = (float*)d_out;

    const size_t N   = NTOK;
    const size_t big = (size_t)BATCH * CQKV * N;   // 37,748,736 floats / buffer

    float* ws     = (float*)d_ws;
    float* qkv    = ws;              // [B,576,N]   (pointwise result)
    float* dwout  = qkv + big;       // [B,576,N]   (depthwise result; q|k|v)
    float* scales = dwout + big;     // [B*384]
    float* accum  = scales + BATCH * 384;              // [B*8,32,32]
    float* attn   = accum + BATCH * HEADS * 32 * 32;   // [B*8,24,24]
    float* avout  = qkv;             // reuse: [B,192,N] (qkv dead after dwconv)

    // 1) qkv pointwise conv (WMMA fp32 GEMM)
    {
        dim3 grid((unsigned)(N / 64), (CQKV / 16) / 4, BATCH);
        gemm_bias_kernel<CQKV, DIM><<<grid, 128, 0, stream>>>(x, qkv_w, qkv_b, qkv);
    }
    // 2) depthwise 3x3x3
    {
        const size_t total = big;
        dwconv_kernel<<<(unsigned)((total + 255) / 256), 256, 0, stream>>>(
            qkv, dw_w, dw_b, dwout);
    }
    // 3) zero Gram accumulators + row norms of q,k
    zero_kernel<<<(BATCH * HEADS * 32 * 32 + 255) / 256, 256, 0, stream>>>(
        accum, BATCH * HEADS * 32 * 32);
    rownorm_kernel<<<BATCH * 384, 256, 0, stream>>>(dwout, scales);
    // 4) q.kT partials (WMMA fp32 + f32 atomics)
    {
        dim3 grid(64, 1, BATCH * HEADS);
        qk_partial_kernel<<<grid, 128, 0, stream>>>(dwout, accum);
    }
    // 5) scale + softmax
    softmax_kernel<<<BATCH * HEADS, 32, 0, stream>>>(accum, scales, temp, attn);
    // 6) attn @ v
    {
        dim3 grid((unsigned)(N / 256), DIM, BATCH);
        av_kernel<<<grid, 256, 0, stream>>>(dwout, attn, avout);
    }
    // 7) projection (WMMA fp32 GEMM) straight into d_out
    {
        dim3 grid((unsigned)(N / 64), (DIM / 16) / 4, BATCH);
        gemm_bias_kernel<DIM, DIM><<<grid, 128, 0, stream>>>(avout, proj_w, proj_b, out);
    }
}